// Semantic_Loss_89386859364662
// MI455X (gfx1250) — compile-verified
//
#include <hip/hip_runtime.h>
#include <hip/hip_bf16.h>

// ---------------------------------------------------------------------------
// Semantic histogram loss for MI455X (gfx1250, wave32).
//   loss = mean(|H(a)-H(p)|) / (mean(|H(a)-H(n)|) + 1e-7)
// where H(x) = 256-bin histogram over channels {0,8,10} of x[-1] (B=8,C=19,H=W=512).
// ---------------------------------------------------------------------------

#define NBINS   256
#define HW      262144          // 512*512
#define HW4     65536           // HW / 4 (float4 elements per channel)
#define NWAVES  8               // 256 threads / wave32
#define SCALE   (256.0f / 255.0f)

typedef __attribute__((ext_vector_type(2))) float v2f;
typedef __attribute__((ext_vector_type(8))) float v8f;

// -------------------------------- zero ws ----------------------------------
__global__ void sl_zero_kernel(unsigned int* __restrict__ gh) {
    int i = blockIdx.x * blockDim.x + threadIdx.x;
    if (i < 3 * NBINS) gh[i] = 0u;
}

// ------------------------------ histogram ----------------------------------
// grid = (64, 1, 3): z selects input tensor; per-wave LDS sub-histograms.
__global__ __launch_bounds__(256) void sl_hist_kernel(
        const float* __restrict__ in0,
        const float* __restrict__ in1,
        const float* __restrict__ in2,
        unsigned int* __restrict__ gh) {
    __shared__ unsigned int lh[NWAVES][NBINS];

    const int tid  = threadIdx.x;
    const int wave = tid >> 5;

    for (int i = tid; i < NWAVES * NBINS; i += 256)
        ((unsigned int*)lh)[i] = 0u;
    __syncthreads();

    const float* in = (blockIdx.z == 0) ? in0 : (blockIdx.z == 1 ? in1 : in2);
    unsigned int* gout = gh + blockIdx.z * NBINS;

    const long base   = 7L * 19L * (long)HW;   // last batch element only
    const int  total4 = 3 * HW4;               // 3 channels of float4s

    for (int e = blockIdx.x * blockDim.x + tid; e < total4;
         e += gridDim.x * blockDim.x) {
        const int c    = e >> 16;              // 0..2 -> channel slab
        const int off4 = e & (HW4 - 1);
        const int chan = (c == 0) ? 0 : (c == 1 ? 8 : 10);
        const float4 v = *(const float4*)(in + base + (long)chan * HW + (long)off4 * 4);

        const float vs[4] = {v.x, v.y, v.z, v.w};
#pragma unroll
        for (int j = 0; j < 4; ++j) {
            const int idx = (int)floorf(vs[j] * SCALE);  // half-open cv2 bins
            if (idx >= 0 && idx < NBINS)                 // validity mask
                atomicAdd(&lh[wave][idx], 1u);
        }
    }
    __syncthreads();

    for (int i = tid; i < NBINS; i += 256) {
        unsigned int s = 0u;
#pragma unroll
        for (int w = 0; w < NWAVES; ++w) s += lh[w][i];
        if (s) atomicAdd(&gout[i], s);
    }
}

// ------------------------------- loss --------------------------------------
// One wave32. The two 256-element L1 reductions run on the matrix pipe:
// four chained V_WMMA_F32_16X16X4_F32 with B = ones produce 16 row-sums;
// column N=0 lives in lanes 0 (rows 0-7) and 16 (rows 8-15).
__global__ __launch_bounds__(32) void sl_loss_kernel(
        const unsigned int* __restrict__ gh, float* __restrict__ out) {
    const unsigned int* ha = gh;
    const unsigned int* hp = gh + NBINS;
    const unsigned int* hn = gh + 2 * NBINS;

    const int l     = threadIdx.x;       // lane 0..31
    const int m     = l & 15;            // A-matrix row
    const int khalf = (l >> 4) * 2;      // A-matrix K pair (0 or 2)

    const v2f ones = {1.0f, 1.0f};
    v8f cap = {};                        // accumulator for |a-p|
    v8f can = {};                        // accumulator for |a-n|

#pragma unroll
    for (int t = 0; t < 4; ++t) {
        // bin -> (tile t, row m, col k): exact 32-bit A-matrix lane layout
        const int b0 = t * 64 + m * 4 + khalf;
        const float a0 = (float)ha[b0];
        const float a1 = (float)ha[b0 + 1];
        v2f Aap = { fabsf(a0 - (float)hp[b0]), fabsf(a1 - (float)hp[b0 + 1]) };
        v2f Aan = { fabsf(a0 - (float)hn[b0]), fabsf(a1 - (float)hn[b0 + 1]) };

        cap = __builtin_amdgcn_wmma_f32_16x16x4_f32(
                false, Aap, false, ones, (short)0, cap, false, false);
        can = __builtin_amdgcn_wmma_f32_16x16x4_f32(
                false, Aan, false, ones, (short)0, can, false, false);
    }

    // per-lane sum of 8 C/D VGPRs = sum of 8 row-sums in column N=lane%16
    float sap = cap[0] + cap[1] + cap[2] + cap[3] + cap[4] + cap[5] + cap[6] + cap[7];
    float san = can[0] + can[1] + can[2] + can[3] + can[4] + can[5] + can[6] + can[7];

    // rows 0-7 at lane 0 (N=0), rows 8-15 at lane 16
    const float tot_ap = __shfl(sap, 0, 32) + __shfl(sap, 16, 32);
    const float tot_an = __shfl(san, 0, 32) + __shfl(san, 16, 32);

    if (l == 0) {
        const float l1_ap = tot_ap * (1.0f / 256.0f);
        const float l1_an = tot_an * (1.0f / 256.0f);
        out[0] = l1_ap / (l1_an + 1e-7f);
    }
}

// ------------------------------ launcher -----------------------------------
extern "C" void kernel_launch(void* const* d_in, const int* in_sizes, int n_in,
                              void* d_out, int out_size, void* d_ws, size_t ws_size,
                              hipStream_t stream) {
    const float* resotored = (const float*)d_in[0];
    const float* positive  = (const float*)d_in[1];
    const float* negative  = (const float*)d_in[2];
    float* out = (float*)d_out;
    unsigned int* gh = (unsigned int*)d_ws;   // 3 * 256 uints

    sl_zero_kernel<<<3, 256, 0, stream>>>(gh);
    sl_hist_kernel<<<dim3(64, 1, 3), 256, 0, stream>>>(resotored, positive, negative, gh);
    sl_loss_kernel<<<1, 32, 0, stream>>>(gh, out);
}